// SAGEConv_38500086841695
// MI455X (gfx1250) — compile-verified
//
#include <hip/hip_runtime.h>
#include <hip/hip_bf16.h>

// SAGEConv fused: gather+mean -> concat -> GEMM(256->128) + bias, fp32 WMMA.
// A-matrix (16x256) per row-tile built in LDS; W (128x256) staged once per
// persistent block in LDS with padded stride for bank-conflict-free b64 reads.
// Edge indices staged in LDS once per tile; output stored non-temporally to
// keep x_feat (102 MB) resident in the 192 MB L2.

typedef float v2f __attribute__((ext_vector_type(2)));
typedef float v8f __attribute__((ext_vector_type(8)));

#define N_OUT_   100000
#define D_IN_    128
#define D2_      256     // 2*D_IN
#define D_OUT_   128
#define TILES_   (N_OUT_ / 16)   // 6250 row-tiles of 16
#define SA_      260             // padded LDS row stride (floats); 260%64=4 -> conflict-free
#define NBLK_    625             // persistent blocks

__global__ __launch_bounds__(256) void sage_fused_wmma(
    const float* __restrict__ x_feat,
    const long long* __restrict__ row_ptr,
    const long long* __restrict__ col_ind,
    const float* __restrict__ W,
    const float* __restrict__ lin_bias,
    const float* __restrict__ add_bias,
    float* __restrict__ out)
{
    __shared__ __align__(16) float xagg[16 * SA_];      // 16 x 256 A-tile (padded)
    __shared__ __align__(16) float wl[D_OUT_ * SA_];    // 128 x 256 W (padded)
    __shared__ int idxbuf[256];                          // 16 rows x 16 edges

    const int t     = threadIdx.x;
    const int lane  = t & 31;
    const int wv    = t >> 5;        // wave 0..7 -> N-tile
    const int laneL = lane & 15;     // 0..15
    const int hi    = lane >> 4;     // 0 or 1 (upper lane half -> K+2 / M+8)

    // ---- stage W into LDS once (coalesced float4 reads) ----
    const float4* W4 = (const float4*)W;
    for (int i = t; i < (D_OUT_ * D2_) / 4; i += 256) {
        float4 f   = W4[i];
        int   row  = i >> 6;          // 64 float4 per 256-col row
        int   c4   = (i & 63) << 2;
        *(float4*)&wl[row * SA_ + c4] = f;
    }

    const int   col  = (wv << 4) + laneL;                 // output column 0..127
    const float bsum = lin_bias[col] + add_bias[col];

    const int r  = t >> 4;          // 0..15 : row within tile (phase 1)
    const int jj = t & 15;          // 0..15 : edge slot within row (phase 1)
    const int cg = jj << 3;         // 0,8,...,120 : 8-float column group (phase 1)

    // per-lane output base (row part added per tile)
    float* outLane = out + (size_t)((hi << 3) * D_OUT_ + col);

    __syncthreads();

    for (int tile = blockIdx.x; tile < TILES_; tile += gridDim.x) {
        const int rowBase = tile << 4;

        // ---- phase 1a: stage edge indices (1 per thread) + self features ----
        long long s = row_ptr[rowBase + r];
        long long e = row_ptr[rowBase + r + 1];
        int deg = (int)(e - s);
        if (jj < deg) {
            long long idx = __builtin_nontemporal_load(&col_ind[s + jj]);
            idxbuf[t] = (int)idx;
        }
        {
            const float4* src = (const float4*)&x_feat[(size_t)(rowBase + r) * D_IN_ + cg];
            float4 a0 = src[0];
            float4 a1 = src[1];
            *(float4*)&xagg[r * SA_ + cg]     = a0;
            *(float4*)&xagg[r * SA_ + cg + 4] = a1;
        }
        // ---- phase 1b: mean over neighbors -> xagg[:, 128:256] ----
        // idxbuf[r*16..] was written by this thread's own 16-lane group (same
        // wave); DS ops are in-order per wave, so no barrier needed.
        {
            float acc0 = 0.f, acc1 = 0.f, acc2 = 0.f, acc3 = 0.f;
            float acc4 = 0.f, acc5 = 0.f, acc6 = 0.f, acc7 = 0.f;
            int dcached = deg < 16 ? deg : 16;
            for (int j = 0; j < dcached; ++j) {
                int idx = idxbuf[(r << 4) + j];
                const float4* src = (const float4*)&x_feat[(size_t)idx * D_IN_ + cg];
                float4 a0 = src[0];
                float4 a1 = src[1];
                acc0 += a0.x; acc1 += a0.y; acc2 += a0.z; acc3 += a0.w;
                acc4 += a1.x; acc5 += a1.y; acc6 += a1.z; acc7 += a1.w;
            }
            for (long long j = s + 16; j < e; ++j) {   // generality tail (dead for DEG=16)
                long long idx = __builtin_nontemporal_load(&col_ind[j]);
                const float4* src = (const float4*)&x_feat[(size_t)idx * D_IN_ + cg];
                float4 a0 = src[0];
                float4 a1 = src[1];
                acc0 += a0.x; acc1 += a0.y; acc2 += a0.z; acc3 += a0.w;
                acc4 += a1.x; acc5 += a1.y; acc6 += a1.z; acc7 += a1.w;
            }
            float fdeg = (float)deg;
            float rdeg = fdeg > 0.f ? 1.f / fdeg : 0.f;
            float4 m0 = make_float4(acc0 * rdeg, acc1 * rdeg, acc2 * rdeg, acc3 * rdeg);
            float4 m1 = make_float4(acc4 * rdeg, acc5 * rdeg, acc6 * rdeg, acc7 * rdeg);
            *(float4*)&xagg[r * SA_ + D_IN_ + cg]     = m0;
            *(float4*)&xagg[r * SA_ + D_IN_ + cg + 4] = m1;
        }
        __syncthreads();

        // ---- phase 2: 16x16 output tile per wave; K=256 in steps of 4 ----
        // A layout (16x4 f32): VGPR0 = (M=laneL, K=2*hi), VGPR1 = (M=laneL, K=2*hi+1)
        // B layout (4x16 f32): VGPR0 = (N=laneL, K=2*hi), VGPR1 = (N=laneL, K=2*hi+1)
        v8f c = {};
        const int aoff = laneL * SA_ + (hi << 1);
        const int boff = col   * SA_ + (hi << 1);   // W row index == output column
        #pragma unroll 8
        for (int k = 0; k < D2_; k += 4) {
            v2f a = *(const v2f*)&xagg[aoff + k];
            v2f b = *(const v2f*)&wl[boff + k];
            c = __builtin_amdgcn_wmma_f32_16x16x4_f32(
                    /*neg_a=*/false, a, /*neg_b=*/false, b,
                    /*c_mod=*/(short)0, c, /*reuse_a=*/false, /*reuse_b=*/false);
        }

        // ---- epilogue: D VGPR v -> (row = rowBase + v + 8*hi, col) ----
        // one base pointer per tile; stores use immediate offsets v*512B.
        {
            float* p = outLane + (size_t)rowBase * D_OUT_;
            #pragma unroll
            for (int v = 0; v < 8; ++v) {
                __builtin_nontemporal_store(c[v] + bsum, &p[v * D_OUT_]);
            }
        }
        __syncthreads();  // protect xagg/idxbuf before next tile's phase 1
    }
}

extern "C" void kernel_launch(void* const* d_in, const int* in_sizes, int n_in,
                              void* d_out, int out_size, void* d_ws, size_t ws_size,
                              hipStream_t stream) {
    const float*     x_feat  = (const float*)d_in[0];
    const long long* row_ptr = (const long long*)d_in[1];
    const long long* col_ind = (const long long*)d_in[2];
    // d_in[3] = sample_count (scalar) — degree derived from row_ptr instead
    const float*     W       = (const float*)d_in[4];
    const float*     lb      = (const float*)d_in[5];
    const float*     bb      = (const float*)d_in[6];
    float*           out     = (float*)d_out;

    sage_fused_wmma<<<NBLK_, 256, 0, stream>>>(x_feat, row_ptr, col_ind, W, lb, bb, out);
}